// EdgeAdaptiveAdj_5463198400723
// MI455X (gfx1250) — compile-verified
//
#include <hip/hip_runtime.h>
#include <hip/hip_bf16.h>

typedef __attribute__((ext_vector_type(2))) float v2f;
typedef __attribute__((ext_vector_type(8))) float v8f;

#define EDGE_EPS 1e-8f

// ---------------------------------------------------------------------------
// Pass 0: zero the per-node row-sum accumulator (d_ws is poisoned by harness).
// ---------------------------------------------------------------------------
__global__ void zero_f32(float* __restrict__ p, int n) {
    int i = blockIdx.x * blockDim.x + threadIdx.x;
    if (i < n) p[i] = 0.0f;
}

// ---------------------------------------------------------------------------
// Pass 1: per 16-edge tile per wave32, compute 16 edge dot-products with
// 4 chained V_WMMA_F32_16X16X4_F32 ops (diag of A x B), sigmoid, store w,
// and atomically accumulate row sums per source node.
//
// f32 16x16x4 operand layout (wave32):
//   A (16x4, MxK): lane L (L<16): V0=A[L][0], V1=A[L][1]; lane L+16: K=2,3
//   B (4x16, KxN): lane j: V0=B[0][j], V1=B[1][j]; lane j+16: K=2,3
// => lane (idx = lane&15, half = lane>>4) loads float2 at dim (4c + 2*half)
//    from emb1[src[e_idx]] (A) and emb2[dst[e_idx]] (B). Same bytes/edge as a
//    scalar kernel -> no bandwidth cost for using the matrix pipe.
//
// D (16x16 f32): VGPR v: lanes 0-15 hold M=v, lanes 16-31 hold M=v+8.
// Diagonal: D[v][v] @ (VGPR v, lane v); D[v+8][v+8] @ (VGPR v, lane v+24).
// ---------------------------------------------------------------------------
__global__ __launch_bounds__(256) void edge_scores_wmma(
    const int*   __restrict__ src,
    const int*   __restrict__ dst,
    const float* __restrict__ emb1,
    const float* __restrict__ emb2,
    float*       __restrict__ w_out,
    float*       __restrict__ row_sum,
    int nTiles) {
    const int lane = threadIdx.x & 31;
    const int tile = blockIdx.x * (blockDim.x >> 5) + (threadIdx.x >> 5);
    if (tile >= nTiles) return;          // wave-uniform branch: EXEC stays all-1s

    const int idx  = lane & 15;          // edge within tile this lane serves
    const int half = lane >> 4;          // which K-half of the row this lane loads
    const int e = tile * 16 + idx;
    const int s = src[e];
    const int d = dst[e];
    const float* __restrict__ rowA = emb1 + (size_t)s * 16 + 2 * half;
    const float* __restrict__ rowB = emb2 + (size_t)d * 16 + 2 * half;

    v8f acc = {};
#pragma unroll
    for (int c = 0; c < 4; ++c) {        // K = 16 in 4 chunks of 4
        v2f a = *(const v2f*)(rowA + 4 * c);
        v2f b = *(const v2f*)(rowB + 4 * c);
        acc = __builtin_amdgcn_wmma_f32_16x16x4_f32(
            /*neg_a=*/false, a, /*neg_b=*/false, b,
            /*c_mod=*/(short)0, acc, /*reuse_a=*/false, /*reuse_b=*/false);
    }

    // Pull the diagonal out of the accumulator tile.
    float diag = 0.0f;
#pragma unroll
    for (int v = 0; v < 8; ++v) {
        bool sel = (lane == v) || (lane == v + 24);
        diag = sel ? acc[v] : diag;
    }
    // Scores for edges 0..7 sit in lanes 0..7; edges 8..15 in lanes 24..31.
    // Compact into lanes 0..15 with a backward permute (byte index).
    int srcLane = (idx < 8) ? idx : (idx + 16);
    float sc = __int_as_float(
        __builtin_amdgcn_ds_bpermute(srcLane << 2, __float_as_int(diag)));

    if (half == 0) {
        float w = 1.0f / (1.0f + __expf(-sc));   // sigmoid
        w_out[e] = w;
        atomicAdd(row_sum + s, w);               // global_atomic_add_f32
    }
}

// ---------------------------------------------------------------------------
// Scalar tail for E % 16 edges (not launched when E is a multiple of 16).
// ---------------------------------------------------------------------------
__global__ void edge_scores_tail(
    const int* __restrict__ src, const int* __restrict__ dst,
    const float* __restrict__ emb1, const float* __restrict__ emb2,
    float* __restrict__ w_out, float* __restrict__ row_sum,
    int start, int E) {
    int e = start + blockIdx.x * blockDim.x + threadIdx.x;
    if (e >= E) return;
    int s = src[e], d = dst[e];
    const float4* a = (const float4*)(emb1 + (size_t)s * 16);
    const float4* b = (const float4*)(emb2 + (size_t)d * 16);
    float sc = 0.0f;
#pragma unroll
    for (int i = 0; i < 4; ++i) {
        float4 av = a[i], bv = b[i];
        sc += av.x * bv.x + av.y * bv.y + av.z * bv.z + av.w * bv.w;
    }
    float w = 1.0f / (1.0f + __expf(-sc));
    w_out[e] = w;
    atomicAdd(row_sum + s, w);
}

// ---------------------------------------------------------------------------
// Pass 2: normalize in place: w[e] /= (row_sum[src[e]] + eps)
// ---------------------------------------------------------------------------
__global__ void normalize_edges(const int* __restrict__ src,
                                const float* __restrict__ row_sum,
                                float* __restrict__ w, int E) {
    int e = blockIdx.x * blockDim.x + threadIdx.x;
    if (e >= E) return;
    w[e] = w[e] / (row_sum[src[e]] + EDGE_EPS);
}

extern "C" void kernel_launch(void* const* d_in, const int* in_sizes, int n_in,
                              void* d_out, int out_size, void* d_ws, size_t ws_size,
                              hipStream_t stream) {
    const int*   src  = (const int*)d_in[0];
    const int*   dst  = (const int*)d_in[1];
    const float* emb1 = (const float*)d_in[2];
    const float* emb2 = (const float*)d_in[3];
    float* w       = (float*)d_out;
    float* row_sum = (float*)d_ws;      // N floats of scratch

    const int E = in_sizes[0];
    const int N = in_sizes[2] / 16;     // EMB_DIM = 16

    // Pass 0: zero row sums (required every call: ws is not re-zeroed).
    zero_f32<<<(N + 255) / 256, 256, 0, stream>>>(row_sum, N);

    // Pass 1: WMMA-tiled scores + sigmoid + atomic row-sum scatter.
    const int nTiles = E / 16;
    if (nTiles > 0) {
        const int wavesPerBlock = 256 / 32;                     // 8 waves/block
        const int blocks = (nTiles + wavesPerBlock - 1) / wavesPerBlock;
        edge_scores_wmma<<<blocks, 256, 0, stream>>>(
            src, dst, emb1, emb2, w, row_sum, nTiles);
    }
    const int rem = E - nTiles * 16;
    if (rem > 0) {
        edge_scores_tail<<<(rem + 255) / 256, 256, 0, stream>>>(
            src, dst, emb1, emb2, w, row_sum, nTiles * 16, E);
    }

    // Pass 2: normalize each edge weight in place.
    normalize_edges<<<(E + 255) / 256, 256, 0, stream>>>(src, row_sum, w, E);
}